// JointTag_withfeature_56040733278608
// MI455X (gfx1250) — compile-verified
//
#include <hip/hip_runtime.h>
#include <hip/hip_bf16.h>
#include <stddef.h>
#include <stdint.h>

// Problem constants (from reference)
#define BB 64
#define LL 512
#define DD 768
#define NF 100
#define WAIST 768
#define NTOK (BB*LL)          // 32768
#define KRAW (DD + NF)        // 868
#define KPAD (DD + 128)       // 896 = 28 * 32
#define NCHUNK (KPAD/32)      // 28
#define TOKL 3
#define CLSL 2

typedef float v2f __attribute__((ext_vector_type(2)));
typedef float v4f __attribute__((ext_vector_type(4)));
typedef float v8f __attribute__((ext_vector_type(8)));
typedef unsigned int u32x4 __attribute__((ext_vector_type(4)));
typedef int i32x4 __attribute__((ext_vector_type(4)));
typedef int i32x8 __attribute__((ext_vector_type(8)));

// ---- TDM availability (device pass only; host pass uses fallback parse) ----
#if defined(__HIP_DEVICE_COMPILE__) && defined(__gfx1250__) && defined(__has_builtin)
#  if __has_builtin(__builtin_amdgcn_tensor_load_to_lds)
#    define USE_TDM 1
#  else
#    define USE_TDM 0
#  endif
#else
#  define USE_TDM 0
#endif

#if __has_include(<hip/amd_detail/amd_gfx1250_TDM.h>)
#  define TDM_SIX_ARGS 1
#else
#  define TDM_SIX_ARGS 0
#endif

// ---------------------------------------------------------------------------
// Kernel 1: build compaction gather index.
// idxFlat[b*L + t] = flat source row of the t-th valid token in batch b, or -1.
// ---------------------------------------------------------------------------
__global__ __launch_bounds__(LL) void build_idx_kernel(const int* __restrict__ valid,
                                                       int* __restrict__ idxFlat) {
  __shared__ int scan[LL];
  __shared__ int srcS[LL];
  const int t = threadIdx.x;
  const int b = blockIdx.x;
  const int v = valid[b * LL + t];
  scan[t] = v;
  __syncthreads();
  for (int off = 1; off < LL; off <<= 1) {
    int val = scan[t] + ((t >= off) ? scan[t - off] : 0);
    __syncthreads();
    scan[t] = val;
    __syncthreads();
  }
  srcS[t] = -1;
  __syncthreads();
  if (v == 1) srcS[scan[t] - 1] = t;
  __syncthreads();
  idxFlat[b * LL + t] = (srcS[t] >= 0) ? (b * LL + srcS[t]) : -1;
}

// ---------------------------------------------------------------------------
// Kernel 2: enriched = relu(features @ enr_w + enr_b), zero-padded to 128 cols.
// ---------------------------------------------------------------------------
#define ENR_TOK 64
__global__ __launch_bounds__(256) void enrich_kernel(const float* __restrict__ feat,
                                                     const float* __restrict__ ew,
                                                     const float* __restrict__ eb,
                                                     float* __restrict__ enr_pad) {
  __shared__ float wS[NF * NF];
  __shared__ float fS[ENR_TOK * NF];
  __shared__ float bS[NF];
  const int tid = threadIdx.x;
  const int tokBase = blockIdx.x * ENR_TOK;
  for (int i = tid; i < NF * NF; i += 256) wS[i] = ew[i];
  for (int i = tid; i < ENR_TOK * NF; i += 256) fS[i] = feat[(size_t)tokBase * NF + i];
  if (tid < NF) bS[tid] = eb[tid];
  __syncthreads();
  for (int o = tid; o < ENR_TOK * 128; o += 256) {
    const int tok = o >> 7;
    const int g = o & 127;
    float v = 0.f;
    if (g < NF) {
      float s = bS[g];
      #pragma unroll 4
      for (int k = 0; k < NF; ++k) s += fS[tok * NF + k] * wS[k * NF + g];
      v = s > 0.f ? s : 0.f;
    }
    enr_pad[(size_t)(tokBase + tok) * 128 + g] = v;
  }
}

// ---------------------------------------------------------------------------
// TDM helper: DMA one 32(k) x 256(n) f32 tile of w1 into LDS.
// tensor_dims are passed as *remaining extent from the tile start* so rows
// past k=867 come back as OOB zeros (TDM coordinate math is tile-local).
// ---------------------------------------------------------------------------
#if USE_TDM
__device__ __forceinline__ void tdm_load_b_tile(const float* __restrict__ w1base,
                                                unsigned ldsAddr, int kg0, int nBase) {
  const unsigned long long ga =
      (unsigned long long)(uintptr_t)(w1base + (size_t)kg0 * WAIST + nBase);
  const unsigned tdim0 = (unsigned)(WAIST - nBase);  // remaining cols
  const unsigned tdim1 = (unsigned)(KRAW - kg0);     // remaining rows (OOB -> 0)
  // D# group 0: count=1 | lds_addr | global_addr[56:0] | type=2
  u32x4 g0 = {1u,
              ldsAddr,
              (unsigned)(ga & 0xFFFFFFFFull),
              (unsigned)((ga >> 32) & 0x01FFFFFFull) | (2u << 30)};
  // D# group 1: data_size=4B; tensor_dim0/1; tile 256x32; dim0 stride = 768
  i32x8 g1 = {(int)(2u << 16),
              (int)((tdim0 & 0xFFFFu) << 16),
              (int)((tdim1 & 0xFFFFu) << 16),
              (int)(256u << 16),
              32,
              (int)WAIST,
              0, 0};
  i32x4 gz = {0, 0, 0, 0};
#if TDM_SIX_ARGS
  i32x8 gz8 = {0, 0, 0, 0, 0, 0, 0, 0};
  __builtin_amdgcn_tensor_load_to_lds(g0, g1, gz, gz, gz8, 0);
#else
  __builtin_amdgcn_tensor_load_to_lds(g0, g1, gz, gz, 0);
#endif
}
#endif

// ---------------------------------------------------------------------------
// Kernel 3: main fused GEMM.
//   X = [compact(seq) | enriched_pad]  (gathered on the fly, never stored)
//   h = relu(X @ w1 + b1)   -> tile kept in LDS only
//   part[nGroup][row][t] = partial h@w2 over this block's 256 cols  (fused)
// Block: 256 threads (8 waves). Output tile 32 rows x 256 cols.
// B tile double-buffered and DMA'd by the Tensor Data Mover (overlapped).
// ---------------------------------------------------------------------------
#define A_STR 36   // padded A row stride (bank-conflict-free, 16B aligned)
__global__ __launch_bounds__(256) void gemm1_fused_kernel(
    const float* __restrict__ seq, const float* __restrict__ enr_pad,
    const int* __restrict__ idxFlat, const float* __restrict__ w1,
    const float* __restrict__ b1, const float* __restrict__ w2,
    float* __restrict__ part) {
  __shared__ float As[32 * A_STR];     // 4.5 KB
  __shared__ float Bs[2][32 * 256];    // 64 KB, ping-pong
  __shared__ int idxS[32];

  const int tid = threadIdx.x;
  const int rowBase = blockIdx.x * 32;
  const int nGroup = blockIdx.y;       // 0..2
  const int nBase = nGroup * 256;

  if (tid < 32) idxS[tid] = idxFlat[rowBase + tid];

  const int wave = tid >> 5;
  const int lane = tid & 31;
  const int laneLo = lane & 15;
  const int laneHi = lane >> 4;
  const int mSub = wave >> 2;          // 0..1
  const int nQuad = wave & 3;          // 0..3

#if USE_TDM
  const unsigned bsAddr0 = (unsigned)(uintptr_t)&Bs[0][0];
  const unsigned bsAddr1 = (unsigned)(uintptr_t)&Bs[1][0];
  if (wave == 0) tdm_load_b_tile(w1, bsAddr0, 0, nBase);   // prefetch chunk 0
#endif

  v8f acc[4];
  #pragma unroll
  for (int nn = 0; nn < 4; ++nn) {
    v8f z = {0.f, 0.f, 0.f, 0.f, 0.f, 0.f, 0.f, 0.f};
    acc[nn] = z;
  }
  const v4f zero4 = {0.f, 0.f, 0.f, 0.f};

  __syncthreads();  // idxS visible

  for (int chunk = 0; chunk < NCHUNK; ++chunk) {
    // ---- stage A tile: 32 rows x 32 k (gathered compaction), float4/thread --
    {
      const int i = tid >> 3;
      const int c4 = (tid & 7) * 4;
      const int kg = chunk * 32 + c4;
      v4f v;
      if (kg < DD) {
        const int s = idxS[i];
        v = (s >= 0) ? *(const v4f*)(seq + (size_t)s * DD + kg) : zero4;
      } else {
        v = *(const v4f*)(enr_pad + (size_t)(rowBase + i) * 128 + (kg - DD));
      }
      *(v4f*)&As[i * A_STR + c4] = v;
    }

#if USE_TDM
    if (wave == 0) __builtin_amdgcn_s_wait_tensorcnt(0);  // current buf landed
    __syncthreads();
    if (wave == 0 && chunk + 1 < NCHUNK)                  // overlap next DMA
      tdm_load_b_tile(w1, ((chunk + 1) & 1) ? bsAddr1 : bsAddr0,
                      (chunk + 1) * 32, nBase);
#else
    // ---- fallback: manual B staging into the ping-pong buffer ----
    {
      float* Bw = &Bs[chunk & 1][0];
      const int col4 = (tid & 63) * 4;
      const int r0 = tid >> 6;
      #pragma unroll
      for (int j = 0; j < 8; ++j) {
        const int rk = j * 4 + r0;
        const int kg = chunk * 32 + rk;
        v4f v = (kg < KRAW) ? *(const v4f*)(w1 + (size_t)kg * WAIST + nBase + col4)
                            : zero4;
        *(v4f*)&Bw[rk * 256 + col4] = v;
      }
    }
    __syncthreads();
#endif

    // ---- 8 K-steps of 4, 4 WMMAs each ----
    const float* Bcur = &Bs[chunk & 1][0];
    #pragma unroll
    for (int k4 = 0; k4 < 8; ++k4) {
      const int k0 = k4 * 4;
      const float* ap = &As[(mSub * 16 + laneLo) * A_STR + k0 + laneHi * 2];
      v2f a;
      a.x = ap[0];
      a.y = ap[1];
      #pragma unroll
      for (int nn = 0; nn < 4; ++nn) {
        const float* bp = &Bcur[(k0 + laneHi * 2) * 256 + nQuad * 64 + nn * 16 + laneLo];
        v2f b;
        b.x = bp[0];
        b.y = bp[256];
        acc[nn] = __builtin_amdgcn_wmma_f32_16x16x4_f32(
            false, a, false, b, (short)0, acc[nn], false, false);
      }
    }
    __syncthreads();
  }

  // ---- epilogue: bias + relu, park h tile in LDS (reuse buffer 0) ----
  float* hS = &Bs[0][0];
  #pragma unroll
  for (int nn = 0; nn < 4; ++nn) {
    const int colL = nQuad * 64 + nn * 16 + laneLo;
    const float bias = b1[nBase + colL];
    const int row0 = mSub * 16 + laneHi * 8;  // C layout: VGPR r -> rows r / r+8
    #pragma unroll
    for (int r = 0; r < 8; ++r) {
      float h = acc[nn][r] + bias;
      h = h > 0.f ? h : 0.f;
      hS[(row0 + r) * 256 + colL] = h;
    }
  }
  __syncthreads();

  // ---- fused skinny GEMM partial: 32 rows x 3 labels over 256 cols ----
  if (tid < 32 * TOKL) {
    const int row = tid / TOKL;
    const int t = tid % TOKL;
    float s = 0.f;
    #pragma unroll 4
    for (int n = 0; n < 256; ++n)
      s += hS[row * 256 + n] * w2[(size_t)(nBase + n) * TOKL + t];
    part[(size_t)nGroup * (NTOK * TOKL) + (size_t)(rowBase + row) * TOKL + t] = s;
  }
}

// ---------------------------------------------------------------------------
// Kernel 4: deterministic reduce of the 3 N-group partials + b2 -> tag_logits
// ---------------------------------------------------------------------------
__global__ __launch_bounds__(256) void tag_reduce_kernel(const float* __restrict__ part,
                                                         const float* __restrict__ b2,
                                                         float* __restrict__ out_tag) {
  const int i = blockIdx.x * 256 + threadIdx.x;
  if (i < NTOK * TOKL) {
    const int t = i % TOKL;
    out_tag[i] = b2[t] + part[i] + part[NTOK * TOKL + i] + part[2 * NTOK * TOKL + i];
  }
}

// ---------------------------------------------------------------------------
// Kernel 5: cls head: cls_logits = seq[:,0,:] @ cls_w + cls_b
// ---------------------------------------------------------------------------
__global__ __launch_bounds__(128) void cls_kernel(const float* __restrict__ seq,
                                                  const float* __restrict__ cw,
                                                  const float* __restrict__ cb,
                                                  float* __restrict__ out) {
  const int t = threadIdx.x;
  const int b = t >> 1;
  const int c = t & 1;
  const float* row = seq + (size_t)b * LL * DD;
  float s = cb[c];
  #pragma unroll 4
  for (int d = 0; d < DD; ++d) s += row[d] * cw[d * CLSL + c];
  out[b * CLSL + c] = s;
}

// ---------------------------------------------------------------------------
extern "C" void kernel_launch(void* const* d_in, const int* in_sizes, int n_in,
                              void* d_out, int out_size, void* d_ws, size_t ws_size,
                              hipStream_t stream) {
  (void)in_sizes; (void)n_in; (void)out_size; (void)ws_size;
  const float* seq   = (const float*)d_in[0];
  const float* feat  = (const float*)d_in[1];
  const int*   valid = (const int*)  d_in[2];
  const float* enr_w = (const float*)d_in[3];
  const float* enr_b = (const float*)d_in[4];
  const float* w1    = (const float*)d_in[5];
  const float* b1    = (const float*)d_in[6];
  const float* w2    = (const float*)d_in[7];
  const float* b2    = (const float*)d_in[8];
  const float* cls_w = (const float*)d_in[9];
  const float* cls_b = (const float*)d_in[10];

  float* out_cls = (float*)d_out;              // [64, 2]
  float* out_tag = (float*)d_out + BB * CLSL;  // [64, 512, 3]

  // Workspace layout (16B-aligned offsets): idxFlat | enr_pad | part (~18 MB)
  char* ws = (char*)d_ws;
  int*   idxFlat = (int*)ws;
  float* enr_pad = (float*)(ws + (size_t)NTOK * sizeof(int));
  float* part    = (float*)(ws + (size_t)NTOK * sizeof(int)
                               + (size_t)NTOK * 128 * sizeof(float));

  build_idx_kernel<<<BB, LL, 0, stream>>>(valid, idxFlat);
  enrich_kernel<<<NTOK / ENR_TOK, 256, 0, stream>>>(feat, enr_w, enr_b, enr_pad);
  dim3 grid(NTOK / 32, WAIST / 256, 1);
  gemm1_fused_kernel<<<grid, 256, 0, stream>>>(seq, enr_pad, idxFlat, w1, b1, w2, part);
  cls_kernel<<<1, 128, 0, stream>>>(seq, cls_w, cls_b, out_cls);
  tag_reduce_kernel<<<(NTOK * TOKL + 255) / 256, 256, 0, stream>>>(part, b2, out_tag);
}